// Anchors_38388417692313
// MI455X (gfx1250) — compile-verified
//
#include <hip/hip_runtime.h>
#include <math.h>

#define NBOX 64
#define A_TOTAL 195840   // (128^2 + 64^2 + 32^2 + 16^2) * 9 for input_size = 512
#define BLOCK 256

struct SMem {
  float4 corner[NBOX];       // (bx1, by1, bx2, by2) -> one ds_load_b128/iter
  float4 xywh[NBOX];         // (cx, cy, w, h) for the final gather
  float  raw[NBOX * 4];      // async-staged boxes (x1,y1,x2,y2)
  float  areab[NBOX];
  int    lab[NBOX];
};

__global__ __launch_bounds__(BLOCK) void anchor_match_kernel(
    const int* __restrict__ labels,
    const float* __restrict__ boxes,
    float* __restrict__ out,
    int B)
{
  __shared__ SMem sm;
  const int t = threadIdx.x;
  const int b = blockIdx.y;

  // ---- Stage this batch's 64 boxes + labels into LDS via CDNA5 async copies
  // (ASYNCcnt-tracked, bypasses VGPRs; waited with s_wait_asynccnt).
  if (t < NBOX) {
    unsigned lraw = (unsigned)(unsigned long long)(void*)(&sm.raw[t * 4]);
    unsigned llab = (unsigned)(unsigned long long)(void*)(&sm.lab[t]);
    unsigned long long graw =
        (unsigned long long)(const void*)(boxes + ((size_t)b * NBOX + t) * 4);
    unsigned long long glab =
        (unsigned long long)(const void*)(labels + (size_t)b * NBOX + t);
    asm volatile("global_load_async_to_lds_b128 %0, %1, off"
                 :: "v"(lraw), "v"(graw) : "memory");
    asm volatile("global_load_async_to_lds_b32 %0, %1, off"
                 :: "v"(llab), "v"(glab) : "memory");
    asm volatile("s_wait_asynccnt 0" ::: "memory");
  }
  __syncthreads();

  // ---- Derive per-box corner form, xywh form, and area (once per block)
  if (t < NBOX) {
    float x1 = sm.raw[t * 4 + 0], y1 = sm.raw[t * 4 + 1];
    float x2 = sm.raw[t * 4 + 2], y2 = sm.raw[t * 4 + 3];
    float cx = (x1 + x2) * 0.5f;             // reference: (a+b)/2.0
    float cy = (y1 + y2) * 0.5f;
    float w  = x2 - x1 + 1.0f;               // reference: b-a+1.0
    float h  = y2 - y1 + 1.0f;
    float bx1 = cx - w * 0.5f, bx2 = cx + w * 0.5f;
    float by1 = cy - h * 0.5f, by2 = cy + h * 0.5f;
    sm.xywh[t]   = make_float4(cx, cy, w, h);
    sm.corner[t] = make_float4(bx1, by1, bx2, by2);
    sm.areab[t]  = (bx2 - bx1 + 1.0f) * (by2 - by1 + 1.0f);
  }
  __syncthreads();

  const int ai = blockIdx.x * BLOCK + t;
  if (ai >= A_TOTAL) return;

  // ---- Reconstruct anchor (cx,cy,w,h) arithmetically from flat index.
  // Level extents: 128^2*9=147456, 64^2*9=36864, 32^2*9=9216, 16^2*9=2304.
  int lvl, r;
  if (ai < 147456)      { lvl = 0; r = ai; }
  else if (ai < 184320) { lvl = 1; r = ai - 147456; }
  else if (ai < 193536) { lvl = 2; r = ai - 184320; }
  else                  { lvl = 3; r = ai - 193536; }
  const int fm_log2 = 7 - lvl;               // fm = 128 >> lvl
  const int k    = r % 9;
  const int cell = r / 9;
  const int col  = cell & ((1 << fm_log2) - 1);
  const int row  = cell >> fm_log2;
  const float gridsz = (float)(4 << lvl);    // 4,8,16,32 (exact in fp32)
  const float acx = ((float)col + 0.5f) * gridsz;
  const float acy = ((float)row + 0.5f) * gridsz;

  // Anchor w/h: double math then fp32 cast, bit-matching numpy's table build.
  const double s_area = (lvl == 0) ? 16.0 : (lvl == 1) ? 256.0
                        : (lvl == 2) ? 4096.0 : 16384.0;
  const int ari = k / 3, sri = k - ari * 3;
  const double ar = (ari == 0) ? 0.5 : (ari == 1) ? 1.0 : 2.0;
  const double sr = (sri == 0) ? 1.0 : (sri == 1) ? 2.0 : 0.75;
  const double hh = sqrt(s_area / ar);
  const double ww = ar * hh;
  const float awf = (float)(ww * sr);
  const float ahf = (float)(hh * sr);

  const float ax1 = acx - awf * 0.5f, ax2 = acx + awf * 0.5f;
  const float ay1 = acy - ahf * 0.5f, ay2 = acy + ahf * 0.5f;
  const float area_a = (ax2 - ax1 + 1.0f) * (ay2 - ay1 + 1.0f);

  // ---- IoU max/argmax over 64 boxes: 1x ds_load_b128 + 1x ds_load_b32 per
  // iteration, LDS broadcast (all lanes read the same box).
  float best = -1.0f;
  int   bi   = 0;
#pragma unroll 4
  for (int n = 0; n < NBOX; ++n) {
    const float4 c4 = sm.corner[n];
    float iw = fminf(ax2, c4.z) - fmaxf(ax1, c4.x) + 1.0f;
    float ih = fminf(ay2, c4.w) - fmaxf(ay1, c4.y) + 1.0f;
    iw = fmaxf(iw, 0.0f);
    ih = fmaxf(ih, 0.0f);
    const float inter = iw * ih;
    const float iou = inter / (area_a + sm.areab[n] - inter);  // precise fdiv
    if (iou > best) { best = iou; bi = n; }   // strict > => first occurrence
  }

  // ---- Targets.
  const float4 m = sm.xywh[bi];              // matched box (cx,cy,w,h)

  int c = sm.lab[bi];
  if (best < 0.5f) c = 0;
  if (best > 0.4f && best < 0.5f) c = -1;

  const size_t idx = (size_t)b * A_TOTAL + (size_t)ai;
  out[idx] = (float)c;                                  // cls (flat, first)

  float* loc = out + (size_t)B * A_TOTAL + idx * 4;     // loc (flat, second)
  loc[0] = (m.x - acx) / awf;
  loc[1] = (m.y - acy) / ahf;
  loc[2] = logf(m.z / awf);
  loc[3] = logf(m.w / ahf);
}

extern "C" void kernel_launch(void* const* d_in, const int* in_sizes, int n_in,
                              void* d_out, int out_size, void* d_ws, size_t ws_size,
                              hipStream_t stream) {
  (void)d_ws; (void)ws_size; (void)out_size; (void)n_in;
  const int*   labels = (const int*)d_in[0];
  const float* boxes  = (const float*)d_in[1];
  float*       out    = (float*)d_out;
  const int B = in_sizes[0] / NBOX;          // labels is [B, 64]
  dim3 grid((A_TOTAL + BLOCK - 1) / BLOCK, B);
  anchor_match_kernel<<<grid, BLOCK, 0, stream>>>(labels, boxes, out, B);
}